// FlexibleSAGEReranker_48885317763297
// MI455X (gfx1250) — compile-verified
//
#include <hip/hip_runtime.h>

// FlexibleSAGEReranker for gfx1250 (MI455X).
// fp32 end-to-end; GEMMs on the matrix pipe via V_WMMA_F32_16X16X4_F32.

typedef __attribute__((ext_vector_type(2))) float v2f;
typedef __attribute__((ext_vector_type(8))) float v8f;

#define DIM 128
#define TILE_M 32            // rows per block
#define SUB (TILE_M / 16)    // 16x16 row sub-tiles per wave
#define LN_EPS 1e-5f

// ---------------------------------------------------------------- degree ----
__global__ __launch_bounds__(256) void k_degree(const int* __restrict__ dst,
                                                float* __restrict__ deg, int e) {
  int i = blockIdx.x * blockDim.x + threadIdx.x;
  if (i < e) atomicAdd(&deg[dst[i]], 1.0f);   // global_atomic_add_f32
}

__global__ __launch_bounds__(256) void k_invdeg(float* __restrict__ deg, int n) {
  int i = blockIdx.x * blockDim.x + threadIdx.x;
  if (i < n) deg[i] = 1.0f / fmaxf(deg[i], 1.0f);
}

// --------------------------------------------------- edge scatter (mean) ----
// One wave32 per edge: lane l moves columns [4l, 4l+4).
__global__ __launch_bounds__(256) void k_scatter(const float* __restrict__ h,
                                                 const int* __restrict__ src,
                                                 const int* __restrict__ dst,
                                                 float* __restrict__ agg, int e) {
  int wave = (int)((blockIdx.x * blockDim.x + threadIdx.x) >> 5);
  int lane = threadIdx.x & 31;
  if (wave >= e) return;
  int s = src[wave];
  int d = dst[wave];
  const float4 v = *(const float4*)(h + (size_t)s * DIM + lane * 4);
  float* o = agg + (size_t)d * DIM + lane * 4;
  atomicAdd(o + 0, v.x);
  atomicAdd(o + 1, v.y);
  atomicAdd(o + 2, v.z);
  atomicAdd(o + 3, v.w);
}

// ------------------------------------------------------ fused SAGE layer ----
// block = 256 threads = 8 waves; 32 output rows per block. Wave w computes
// cols [16w,16w+16) for both 16-row sub-tiles, reusing B fragments.
// h_new = relu(agg@Wl^T + bl + h@Wr^T); LayerNorm; + residual.
// IS_FIRST: residual = x@Wp^T + bp fused as a third WMMA accumulator chain.
template <int IS_FIRST>
__global__ __launch_bounds__(256) void k_layer(
    const float* __restrict__ h_in,
    const float* __restrict__ agg,
    const float* __restrict__ invdeg,
    const float* __restrict__ Wl, const float* __restrict__ bl,
    const float* __restrict__ Wr,
    const float* __restrict__ lng, const float* __restrict__ lnb,
    const float* __restrict__ Wp, const float* __restrict__ bp,
    float* __restrict__ h_out, int n) {
  __shared__ float sH[TILE_M * DIM];   // h tile (A operand; residual L1/L2)
  __shared__ float sA[TILE_M * DIM];   // agg tile (A operand) -> relu'd h_new
  __shared__ float sSum[256], sSq[256];
  __shared__ float sMu[TILE_M], sRs[TILE_M];

  const int t = threadIdx.x;
  const int r0 = blockIdx.x * TILE_M;

  // Stage h & agg tiles into LDS (float4, coalesced); fold in 1/deg.
  for (int i = t; i < TILE_M * DIM / 4; i += 256) {
    int row = (i * 4) / DIM;
    int grow = r0 + row;
    if (grow >= n) grow = n - 1;                  // clamp (N % 32 == 0 anyway)
    size_t g = (size_t)grow * DIM + ((i * 4) % DIM);
    float4 hv = *(const float4*)(h_in + g);
    float4 av = *(const float4*)(agg + g);
    float id = invdeg[grow];
    av.x *= id; av.y *= id; av.z *= id; av.w *= id;
    *(float4*)(sH + i * 4) = hv;
    *(float4*)(sA + i * 4) = av;
  }
  __syncthreads();

  const int wave = t >> 5;
  const int lane = t & 31;
  const int hl   = lane >> 4;      // K-half selector for 16x16x4 fragments
  const int m    = lane & 15;      // A-row / B-col within a 16x16 tile
  const int wcol = wave * 16 + m;  // output column this lane produces

  const float* wl_row = Wl + (size_t)wcol * DIM;
  const float* wr_row = Wr + (size_t)wcol * DIM;
  const float* wp_row = Wp + (size_t)wcol * DIM;

  v8f acc[SUB] = {};
  v8f accR[SUB] = {};

#pragma unroll 4
  for (int k0 = 0; k0 < DIM; k0 += 4) {
    const int ka = k0 + 2 * hl;                 // even -> 8B-aligned loads
    const v2f bL = *(const v2f*)(wl_row + ka);
    const v2f bR = *(const v2f*)(wr_row + ka);
    v2f bP = {0.f, 0.f};
    if (IS_FIRST) bP = *(const v2f*)(wp_row + ka);
#pragma unroll
    for (int st = 0; st < SUB; ++st) {
      const v2f aA = *(const v2f*)(sA + (st * 16 + m) * DIM + ka);
      acc[st] = __builtin_amdgcn_wmma_f32_16x16x4_f32(false, aA, false, bL,
                                                      (short)0, acc[st],
                                                      false, false);
      const v2f aH = *(const v2f*)(sH + (st * 16 + m) * DIM + ka);
      acc[st] = __builtin_amdgcn_wmma_f32_16x16x4_f32(false, aH, false, bR,
                                                      (short)0, acc[st],
                                                      false, false);
      if (IS_FIRST)
        accR[st] = __builtin_amdgcn_wmma_f32_16x16x4_f32(false, aH, false, bP,
                                                         (short)0, accR[st],
                                                         false, false);
    }
  }
  __syncthreads();   // all waves done reading sA; safe to reuse as h_new tile

  // Bias + ReLU in registers; stage relu'd tile into (dead) sA for LN stats.
  const float bias = bl[wcol];
#pragma unroll
  for (int st = 0; st < SUB; ++st) {
#pragma unroll
    for (int j = 0; j < 8; ++j) {              // C layout: VGPR j -> row j+8*hl
      const int row = st * 16 + j + 8 * hl;
      const float v = fmaxf(acc[st][j] + bias, 0.0f);
      acc[st][j] = v;
      sA[row * DIM + wcol] = v;
    }
  }
  __syncthreads();

  // LayerNorm statistics: 8 threads per row, 16 elements each.
  {
    const int row = t >> 3, sg = t & 7;
    const float* p = sA + row * DIM + sg * 16;
    float s = 0.f, q = 0.f;
#pragma unroll
    for (int k = 0; k < 16; ++k) { float v = p[k]; s += v; q += v * v; }
    sSum[t] = s; sSq[t] = q;
  }
  __syncthreads();
  if (t < TILE_M) {
    float s = 0.f, q = 0.f;
#pragma unroll
    for (int k = 0; k < 8; ++k) { s += sSum[t * 8 + k]; q += sSq[t * 8 + k]; }
    const float mu = s * (1.0f / DIM);
    const float var = q * (1.0f / DIM) - mu * mu;
    sMu[t] = mu;
    sRs[t] = rsqrtf(var + LN_EPS);
  }
  __syncthreads();

  // Normalize register-resident values + residual, store straight to global.
  const float g_c = lng[wcol];
  const float b_c = lnb[wcol];
  const float pbias = IS_FIRST ? bp[wcol] : 0.0f;
#pragma unroll
  for (int st = 0; st < SUB; ++st) {
#pragma unroll
    for (int j = 0; j < 8; ++j) {
      const int row = st * 16 + j + 8 * hl;
      const float res =
          IS_FIRST ? (accR[st][j] + pbias) : sH[row * DIM + wcol];
      const float v =
          (acc[st][j] - sMu[row]) * sRs[row] * g_c + b_c + res;
      if (r0 + row < n) h_out[(size_t)(r0 + row) * DIM + wcol] = v;
    }
  }
}

// ------------------------------------------------------------------ head ----
// One wave32 per row: lane float4 partial dot + shuffle reduce.
__global__ __launch_bounds__(256) void k_head(const float* __restrict__ h,
                                              const float* __restrict__ rr,
                                              const float* __restrict__ hw,
                                              const float* __restrict__ hb,
                                              const float* __restrict__ alpha,
                                              float* __restrict__ out, int n) {
  int row = (int)((blockIdx.x * blockDim.x + threadIdx.x) >> 5);
  int lane = threadIdx.x & 31;
  if (row >= n) return;
  float4 v = *(const float4*)(h + (size_t)row * DIM + lane * 4);
  float4 w = *(const float4*)(hw + lane * 4);
  float s = v.x * w.x + v.y * w.y + v.z * w.z + v.w * w.w;
  for (int o = 16; o > 0; o >>= 1) s += __shfl_down(s, o, 32);
  if (lane == 0) {
    float a = 1.0f / (1.0f + __expf(-alpha[0]));
    out[row] = a * rr[row] + (1.0f - a) * (s + hb[0]);
  }
}

// ---------------------------------------------------------------- launch ----
extern "C" void kernel_launch(void* const* d_in, const int* in_sizes, int n_in,
                              void* d_out, int out_size, void* d_ws, size_t ws_size,
                              hipStream_t stream) {
  const float* x      = (const float*)d_in[0];
  const int*   ei     = (const int*)d_in[1];
  const float* rr     = (const float*)d_in[2];
  const float* proj_w = (const float*)d_in[3];
  const float* proj_b = (const float*)d_in[4];
  const float* ll_w   = (const float*)d_in[5];
  const float* ll_b   = (const float*)d_in[6];
  const float* lr_w   = (const float*)d_in[7];
  const float* ln_g   = (const float*)d_in[8];
  const float* ln_b   = (const float*)d_in[9];
  const float* head_w = (const float*)d_in[10];
  const float* head_b = (const float*)d_in[11];
  const float* alpha  = (const float*)d_in[12];
  float* out = (float*)d_out;

  const int N_ = in_sizes[0] / DIM;   // 100000
  const int E_ = in_sizes[1] / 2;     // 1600000
  const int* src = ei;
  const int* dst = ei + E_;

  // workspace: deg[N] | agg[N*D] | h[N*D]  (~103 MB)
  float* deg = (float*)d_ws;
  float* agg = deg + (((size_t)N_ + 255) & ~(size_t)255);
  float* h   = agg + (size_t)N_ * DIM;

  hipMemsetAsync(deg, 0, sizeof(float) * (size_t)N_, stream);
  k_degree<<<(E_ + 255) / 256, 256, 0, stream>>>(dst, deg, E_);
  k_invdeg<<<(N_ + 255) / 256, 256, 0, stream>>>(deg, N_);

  const int layer_blocks = (N_ + TILE_M - 1) / TILE_M;
  for (int i = 0; i < 3; ++i) {
    const float* hin = (i == 0) ? x : h;
    hipMemsetAsync(agg, 0, sizeof(float) * (size_t)N_ * DIM, stream);
    k_scatter<<<(int)(((size_t)E_ * 32 + 255) / 256), 256, 0, stream>>>(
        hin, src, dst, agg, E_);
    if (i == 0) {
      k_layer<1><<<layer_blocks, 256, 0, stream>>>(
          hin, agg, deg,
          ll_w, ll_b, lr_w, ln_g, ln_b, proj_w, proj_b, h, N_);
    } else {
      k_layer<0><<<layer_blocks, 256, 0, stream>>>(
          hin, agg, deg,
          ll_w + (size_t)i * DIM * DIM, ll_b + (size_t)i * DIM,
          lr_w + (size_t)i * DIM * DIM,
          ln_g + (size_t)i * DIM, ln_b + (size_t)i * DIM,
          proj_w, proj_b, h, N_);
    }
  }
  k_head<<<(int)(((size_t)N_ * 32 + 255) / 256), 256, 0, stream>>>(
      h, rr, head_w, head_b, alpha, out, N_);
}